// Model_38946763440736
// MI455X (gfx1250) — compile-verified
//
#include <hip/hip_runtime.h>
#include <hip/hip_bf16.h>
#include <math.h>

#define L_SEQ 4096
#define VOCAB 50257
#define H1 30
#define H2 50
#define G1 (4*H1)   // 120
#define G2 (4*H2)   // 200

typedef float v2f __attribute__((ext_vector_type(2)));
typedef float v8f __attribute__((ext_vector_type(8)));

__device__ __forceinline__ float sigmoidf_(float x) { return 1.0f / (1.0f + expf(-x)); }

// ---------------------------------------------------------------------------
// Kernel 1: embedding gather + input-gate precompute for LSTM1
// gx1[t,g] = sum_j emb[x[t],j] * w_ih1[g,j] + b_ih1[g] + b_hh1[g]
// ---------------------------------------------------------------------------
__global__ void k_embed_gx1(const int* __restrict__ x,
                            const float* __restrict__ emb,
                            const float* __restrict__ w_ih1,
                            const float* __restrict__ b_ih1,
                            const float* __restrict__ b_hh1,
                            float* __restrict__ gx1) {
    __shared__ float e[H1];
    const int t = blockIdx.x;
    const int tid = threadIdx.x;
    const int row = x[t];
    if (tid < H1) e[tid] = emb[(size_t)row * H1 + tid];
    __syncthreads();
    if (tid < G1) {
        float acc = b_ih1[tid] + b_hh1[tid];
        const float* w = w_ih1 + tid * H1;
        #pragma unroll
        for (int j = 0; j < H1; ++j) acc += w[j] * e[j];
        gx1[t * G1 + tid] = acc;
    }
}

// ---------------------------------------------------------------------------
// Kernel 2: LSTM1 sequential scan, single workgroup (128 thr = 4 waves).
// Thread g (<120) keeps w_hh1 row in registers; h broadcast via LDS.
// ---------------------------------------------------------------------------
__global__ void k_lstm1(const float* __restrict__ gx1,
                        const float* __restrict__ w_hh1,
                        float* __restrict__ h1s) {
    __shared__ float h[H1];
    __shared__ float gates[G1];
    const int g = threadIdx.x;
    float wrow[H1];
    if (g < G1) {
        #pragma unroll
        for (int j = 0; j < H1; ++j) wrow[j] = w_hh1[g * H1 + j];
    }
    float c = 0.0f;
    if (g < H1) h[g] = 0.0f;
    __syncthreads();
    for (int t = 0; t < L_SEQ; ++t) {
        if (g < G1) {
            float acc = gx1[t * G1 + g];
            #pragma unroll
            for (int j = 0; j < H1; ++j) acc += wrow[j] * h[j];
            gates[g] = acc;
        }
        __syncthreads();
        if (g < H1) {
            float iv = sigmoidf_(gates[g]);
            float fv = sigmoidf_(gates[H1 + g]);
            float gv = tanhf(gates[2 * H1 + g]);
            float ov = sigmoidf_(gates[3 * H1 + g]);
            c = fv * c + iv * gv;
            float hv = ov * tanhf(c);
            h[g] = hv;
            h1s[t * H1 + g] = hv;
        }
        __syncthreads();
    }
}

// ---------------------------------------------------------------------------
// Kernel 3: x2 = h1 @ w1.T + bl1 ; gx2 = x2 @ w_ih2.T + b_ih2 + b_hh2  (fused)
// ---------------------------------------------------------------------------
__global__ void k_lin1_gx2(const float* __restrict__ h1s,
                           const float* __restrict__ w1,
                           const float* __restrict__ bl1,
                           const float* __restrict__ w_ih2,
                           const float* __restrict__ b_ih2,
                           const float* __restrict__ b_hh2,
                           float* __restrict__ gx2) {
    __shared__ float hr[H1];
    __shared__ float x2[H2];
    const int t = blockIdx.x;
    const int tid = threadIdx.x;
    if (tid < H1) hr[tid] = h1s[t * H1 + tid];
    __syncthreads();
    if (tid < H2) {
        float acc = bl1[tid];
        #pragma unroll
        for (int j = 0; j < H1; ++j) acc += w1[tid * H1 + j] * hr[j];
        x2[tid] = acc;
    }
    __syncthreads();
    if (tid < G2) {
        float acc = b_ih2[tid] + b_hh2[tid];
        #pragma unroll
        for (int j = 0; j < H2; ++j) acc += w_ih2[tid * H2 + j] * x2[j];
        gx2[t * G2 + tid] = acc;
    }
}

// ---------------------------------------------------------------------------
// Kernel 4: LSTM2 sequential scan, single workgroup (256 thr = 8 waves).
// ---------------------------------------------------------------------------
__global__ void k_lstm2(const float* __restrict__ gx2,
                        const float* __restrict__ w_hh2,
                        float* __restrict__ h2s) {
    __shared__ float h[H2];
    __shared__ float gates[G2];
    const int g = threadIdx.x;
    float wrow[H2];
    if (g < G2) {
        #pragma unroll
        for (int j = 0; j < H2; ++j) wrow[j] = w_hh2[g * H2 + j];
    }
    float c = 0.0f;
    if (g < H2) h[g] = 0.0f;
    __syncthreads();
    for (int t = 0; t < L_SEQ; ++t) {
        if (g < G2) {
            float acc = gx2[t * G2 + g];
            #pragma unroll
            for (int j = 0; j < H2; ++j) acc += wrow[j] * h[j];
            gates[g] = acc;
        }
        __syncthreads();
        if (g < H2) {
            float iv = sigmoidf_(gates[g]);
            float fv = sigmoidf_(gates[H2 + g]);
            float gv = tanhf(gates[2 * H2 + g]);
            float ov = sigmoidf_(gates[3 * H2 + g]);
            c = fv * c + iv * gv;
            float hv = ov * tanhf(c);
            h[g] = hv;
            h2s[t * H2 + g] = hv;
        }
        __syncthreads();
    }
}

// ---------------------------------------------------------------------------
// Kernel 5: out[L,V] = h2s[L,50] @ w2.T[50,V] + bl2 via V_WMMA_F32_16X16X4_F32.
// Output-write bound (823 MB @ 23.3 TB/s ≈ 35 us), so fp32 WMMA keeps full
// precision for free. One wave = one 16x16 tile; 4 waves per block.
//   A (16x4 f32): lane<16 holds K=k0,k0+1 ; lane>=16 holds K=k0+2,k0+3.
//   B (4x16 f32): mirrored by lane-half.  C/D: VGPR r -> M = r + 8*(lane>=16).
// Edge handling with NO per-load branches:
//   - N edge: clamp column for loads (always in-bounds), guard only the store.
//   - K  tail (50 = 12*4 + 2): 12 unguarded chunks, then one tail chunk where
//     the half=1 lanes (logical K=50,51) zero their A fragment via cndmask;
//     zero A kills those products, so B needs no masking.
// Output uses non-temporal stores so the 823 MB stream doesn't evict w2
// (10 MB, heavily reused) from the 192 MB L2.
// ---------------------------------------------------------------------------
__global__ void k_out_gemm(const float* __restrict__ h2s,
                           const float* __restrict__ w2,
                           const float* __restrict__ bl2,
                           float* __restrict__ out) {
    const int wave = threadIdx.x >> 5;
    const int lane = threadIdx.x & 31;
    const int NT = (VOCAB + 15) / 16;           // 3142 N-tiles
    const int ntile = blockIdx.x * 4 + wave;
    if (ntile >= NT) return;                    // uniform per wave
    const int n0 = ntile * 16;
    const int m0 = blockIdx.y * 16;

    const int half = lane >> 4;                 // 0 or 1
    const int l = lane & 15;
    const int arow  = m0 + l;                   // M index this lane loads
    const int bcol  = n0 + l;                   // N index this lane covers
    const bool colok = (bcol < VOCAB);
    const int bcolc = colok ? bcol : (VOCAB - 1);  // clamped: loads always valid

    const float* __restrict__ aptr = h2s + (size_t)arow * H2;
    const float* __restrict__ bptr = w2 + (size_t)bcolc * H2;

    v8f acc = {};
    // 12 full K-chunks (K = 0..47), unguarded b64 loads
    #pragma unroll
    for (int kc = 0; kc < 12; ++kc) {
        const int kk = kc * 4 + half * 2;       // 8B-aligned (kk even, row 200B)
        v2f a = *(const v2f*)(aptr + kk);
        v2f b = *(const v2f*)(bptr + kk);
        acc = __builtin_amdgcn_wmma_f32_16x16x4_f32(
            false, a, false, b, (short)0, acc, false, false);
    }
    // tail chunk: logical K = 48..51; only K=48,49 exist
    {
        v2f al = *(const v2f*)(aptr + 48);      // valid for both halves
        v2f b  = *(const v2f*)(bptr + 48);      // half=1 garbage x 0 = 0
        v2f a;
        a.x = half ? 0.0f : al.x;               // v_cndmask, no branch
        a.y = half ? 0.0f : al.y;
        acc = __builtin_amdgcn_wmma_f32_16x16x4_f32(
            false, a, false, b, (short)0, acc, false, false);
    }

    if (colok) {                                // single exec region for epilogue
        const float bias = bl2[bcol];
        #pragma unroll
        for (int r = 0; r < 8; ++r) {
            const int m = m0 + r + half * 8;
            __builtin_nontemporal_store(acc[r] + bias,
                                        out + (size_t)m * VOCAB + bcol);
        }
    }
}

// ---------------------------------------------------------------------------
extern "C" void kernel_launch(void* const* d_in, const int* in_sizes, int n_in,
                              void* d_out, int out_size, void* d_ws, size_t ws_size,
                              hipStream_t stream) {
    const int*   x     = (const int*)  d_in[0];
    const float* emb   = (const float*)d_in[1];
    const float* w_ih1 = (const float*)d_in[2];
    const float* w_hh1 = (const float*)d_in[3];
    const float* b_ih1 = (const float*)d_in[4];
    const float* b_hh1 = (const float*)d_in[5];
    const float* w1    = (const float*)d_in[6];
    const float* bl1   = (const float*)d_in[7];
    const float* w_ih2 = (const float*)d_in[8];
    const float* w_hh2 = (const float*)d_in[9];
    const float* b_ih2 = (const float*)d_in[10];
    const float* b_hh2 = (const float*)d_in[11];
    const float* w2    = (const float*)d_in[12];
    const float* bl2   = (const float*)d_in[13];
    float* out = (float*)d_out;

    // Workspace carve-up (all fp32): 4096*(120+30+200+50) = 6.55 MB
    float* gx1 = (float*)d_ws;
    float* h1s = gx1 + (size_t)L_SEQ * G1;
    float* gx2 = h1s + (size_t)L_SEQ * H1;
    float* h2s = gx2 + (size_t)L_SEQ * G2;

    // 1) embedding + LSTM1 input gates (parallel over L)
    k_embed_gx1<<<L_SEQ, 128, 0, stream>>>(x, emb, w_ih1, b_ih1, b_hh1, gx1);
    // 2) LSTM1 recurrence (serial, single workgroup)
    k_lstm1<<<1, 128, 0, stream>>>(gx1, w_hh1, h1s);
    // 3) linear 30->50 fused with LSTM2 input gates (parallel over L)
    k_lin1_gx2<<<L_SEQ, 256, 0, stream>>>(h1s, w1, bl1, w_ih2, b_ih2, b_hh2, gx2);
    // 4) LSTM2 recurrence (serial, single workgroup)
    k_lstm2<<<1, 256, 0, stream>>>(gx2, w_hh2, h2s);
    // 5) big projection with fp32 WMMA: grid = (ceil(3142/4), 4096/16)
    dim3 grid((( (VOCAB + 15) / 16 ) + 3) / 4, L_SEQ / 16, 1);
    k_out_gemm<<<grid, 128, 0, stream>>>(h2s, w2, bl2, out);
}